// SoftNeighSuperpixelAttn_85117661872428
// MI455X (gfx1250) — compile-verified
//
#include <hip/hip_runtime.h>
#include <math.h>
#include <stdint.h>

typedef __attribute__((ext_vector_type(2))) float v2f;
typedef __attribute__((ext_vector_type(8))) float v8f;

#define B_   2
#define H_   128
#define W_   128
#define C_   512
#define G_   4
#define HD_  128
#define KS_  7
#define R_   3
#define QKN  (2 * C_)          // 1024 columns of the projection
#define HWK  (H_ * W_)         // 16384
#define NPIX (B_ * HWK)        // 32768

static __device__ __forceinline__ int iclip(int v, int lo, int hi) {
  return v < lo ? lo : (v > hi ? hi : v);
}

// CDNA5 async copy: global -> LDS with no VGPR round trip (ASYNCcnt tracked).
// LDS operand is the low 32 bits of the generic shared pointer (== LDS byte
// offset per ISA 10.2 aperture rules). 16B alignment required on both sides.
static __device__ __forceinline__ void async_g2l_b128(void* lds,
                                                      const void* gptr) {
  uint32_t off = (uint32_t)(uintptr_t)lds;
  asm volatile("global_load_async_to_lds_b128 %0, %1, off"
               :: "v"(off), "v"(gptr)
               : "memory");
}
static __device__ __forceinline__ void wait_async0() {
  asm volatile("s_wait_asynccnt 0x0" ::: "memory");
}

// ============================================================================
// Kernel A: qk = x @ W_qk + b   (M=32768, N=1024, K=512), f32 WMMA 16x16x4.
// Block = 256 threads = 8 waves; block tile 64x64, K-chunk 32.
// Staging via global_load_async_to_lds_b128; wave (wm,wn) owns a 16x32 strip.
// ============================================================================
#define BM 64
#define BN 64
#define BK 32
#define LDA 40  // As row stride (floats); 160B keeps every 16B chunk aligned
#define BNS 64  // Bs row stride (floats), [k][n] layout (async-copied as-is)

__global__ __launch_bounds__(256) void qk_gemm_kernel(
    const float* __restrict__ x, const float* __restrict__ Wq,
    const float* __restrict__ bias, float* __restrict__ qk) {
  __shared__ float As[BM * LDA];   // [m][k], padded rows
  __shared__ float Bs[BK * BNS];   // [k][n], natural W_qk layout
  const int tid  = threadIdx.x;
  const int lane = tid & 31;
  const int wid  = tid >> 5;
  const int wm   = wid & 3;
  const int wn   = wid >> 2;
  const int la   = lane & 15;
  const int lh   = lane >> 4;
  const int m0   = blockIdx.y * BM;
  const int n0   = blockIdx.x * BN;

  // Per-thread async staging coordinates (fixed across K-chunks)
  const int amm = tid >> 3;          // 0..31 : A row
  const int akk = (tid & 7) * 4;     // 16B column chunk
  void* aDst0 = &As[amm * LDA + akk];
  void* aDst1 = &As[(amm + 32) * LDA + akk];
  const int bkk = tid >> 3;          // 0..31 : B (k) row
  const int bnn = (tid & 7) * 8;     // two 16B chunks per thread
  void* bDst0 = &Bs[bkk * BNS + bnn];
  void* bDst1 = &Bs[bkk * BNS + bnn + 4];

  v8f acc0 = {}, acc1 = {};

  for (int k0 = 0; k0 < C_; k0 += BK) {
    // Async DMA the 64x32 A tile and 32x64 B tile straight into LDS
    async_g2l_b128(aDst0, &x[(size_t)(m0 + amm) * C_ + k0 + akk]);
    async_g2l_b128(aDst1, &x[(size_t)(m0 + amm + 32) * C_ + k0 + akk]);
    async_g2l_b128(bDst0, &Wq[(size_t)(k0 + bkk) * QKN + n0 + bnn]);
    async_g2l_b128(bDst1, &Wq[(size_t)(k0 + bkk) * QKN + n0 + bnn + 4]);
    wait_async0();
    __syncthreads();

#pragma unroll
    for (int kc = 0; kc < BK; kc += 4) {
      const int ka = kc + lh * 2;  // this half-wave's K pair start
      v2f a = *(const v2f*)&As[(wm * 16 + la) * LDA + ka];
      v2f b0, b1;  // two row-strided reads -> ds_load_2addr_b32
      b0[0] = Bs[ka * BNS + wn * 32 + la];
      b0[1] = Bs[(ka + 1) * BNS + wn * 32 + la];
      b1[0] = Bs[ka * BNS + wn * 32 + 16 + la];
      b1[1] = Bs[(ka + 1) * BNS + wn * 32 + 16 + la];
      acc0 = __builtin_amdgcn_wmma_f32_16x16x4_f32(false, a, false, b0,
                                                   (short)0, acc0, false, false);
      acc1 = __builtin_amdgcn_wmma_f32_16x16x4_f32(false, a, false, b1,
                                                   (short)0, acc1, false, false);
    }
    __syncthreads();
  }

  // Epilogue: C layout — vgpr v: lanes 0-15 -> M=v, N=lane; lanes16-31 -> M=v+8
  const int   nA = n0 + wn * 32 + la;
  const int   nB = nA + 16;
  const float bA = bias[nA];
  const float bB = bias[nB];
  const int   mb = m0 + wm * 16 + lh * 8;
#pragma unroll
  for (int v = 0; v < 8; ++v) {
    qk[(size_t)(mb + v) * QKN + nA] = acc0[v] + bA;
    qk[(size_t)(mb + v) * QKN + nB] = acc1[v] + bB;
  }
}

// ============================================================================
// Kernel B: logp[b][si][h][w][ki] = log(sum_m sims_j[m]*[sinds_j[m]==s_i[si]])
// One thread per (b,h,w,ki).
// ============================================================================
__global__ __launch_bounds__(256) void logp_kernel(
    const float* __restrict__ sims, const int* __restrict__ sinds,
    float* __restrict__ lp) {
  int idx = blockIdx.x * 256 + threadIdx.x;  // exactly B*H*W*49 threads
  const int ki = idx % 49;  idx /= 49;
  const int w  = idx & 127;
  const int h  = (idx >> 7) & 127;
  const int b  = idx >> 14;
  const int dh = ki / KS_, dw = ki % KS_;
  const int hj = iclip(h - R_, 0, H_ - KS_) + dh;
  const int wj = iclip(w - R_, 0, W_ - KS_) + dw;

  const int*   si = &sinds[((size_t)b * HWK + h * W_ + w) * 9];
  const int*   sj = &sinds[((size_t)b * HWK + hj * W_ + wj) * 9];
  const float* pj = &sims [((size_t)b * HWK + hj * W_ + wj) * 9];

  int sjv[9]; float pv[9];
#pragma unroll
  for (int m = 0; m < 9; ++m) { sjv[m] = sj[m]; pv[m] = pj[m]; }

#pragma unroll
  for (int s = 0; s < 9; ++s) {
    const int tgt = si[s];
    float acc = 0.f;
#pragma unroll
    for (int m = 0; m < 9; ++m) acc += (sjv[m] == tgt) ? pv[m] : 0.f;
    const float out = (acc > 0.f) ? logf(fmaxf(acc, 1e-30f)) : -__builtin_inff();
    lp[((size_t)(b * 9 + s) * HWK + (size_t)h * W_ + w) * 49 + ki] = out;
  }
}

// ============================================================================
// Kernel C: banded q.k dots via f32 WMMA + fuse logp + write attn.
// One wave per (b,g,h,w0-tile,dh): 2 WMMA tiles (16 queries x 32 key cols,
// K=128), band select from LDS, 16*9*7 = 1008 output elements per wave.
// ============================================================================
__global__ __launch_bounds__(256) void attn_kernel(
    const float* __restrict__ qk, const float* __restrict__ logp,
    float* __restrict__ attn) {
  __shared__ float Ds[8][16 * 32];
  const int lane = threadIdx.x & 31;
  const int wid  = threadIdx.x >> 5;
  const int la   = lane & 15;
  const int lh   = lane >> 4;

  int idx = blockIdx.x * 8 + wid;       // 0..57343 = B*G*H*(W/16)*7
  const int dh = idx % 7;      idx /= 7;
  const int w0 = (idx & 7) * 16; idx >>= 3;
  const int h  = idx & 127;    idx >>= 7;
  const int g  = idx & 3;
  const int b  = idx >> 2;

  const int hj = iclip(h - R_, 0, H_ - KS_) + dh;
  const int c0 = iclip(w0 - R_, 0, W_ - KS_);

  // A frag: query rows w0..w0+15 of row h; B frags: key cols c0..c0+31 (clamped)
  const size_t qbase =
      ((size_t)(b * HWK + h * W_ + (w0 + la))) * QKN + g * HD_ + lh * 2;
  const int cc0 = (c0 + la      > W_ - 1) ? W_ - 1 : c0 + la;
  const int cc1 = (c0 + 16 + la > W_ - 1) ? W_ - 1 : c0 + 16 + la;
  const size_t kbase0 =
      ((size_t)(b * HWK + hj * W_ + cc0)) * QKN + C_ + g * HD_ + lh * 2;
  const size_t kbase1 =
      ((size_t)(b * HWK + hj * W_ + cc1)) * QKN + C_ + g * HD_ + lh * 2;

  v8f acc0 = {}, acc1 = {};
#pragma unroll
  for (int kc = 0; kc < HD_; kc += 4) {
    v2f a  = *(const v2f*)&qk[qbase  + kc];
    v2f b0 = *(const v2f*)&qk[kbase0 + kc];
    v2f b1 = *(const v2f*)&qk[kbase1 + kc];
    acc0 = __builtin_amdgcn_wmma_f32_16x16x4_f32(false, a, false, b0,
                                                 (short)0, acc0, false, false);
    acc1 = __builtin_amdgcn_wmma_f32_16x16x4_f32(false, a, false, b1,
                                                 (short)0, acc1, false, false);
  }

  // Spill the 16x32 dot band to LDS for band-diagonal selection
  float* D = Ds[wid];
#pragma unroll
  for (int v = 0; v < 8; ++v) {
    D[(v + lh * 8) * 32 + la]      = acc0[v];
    D[(v + lh * 8) * 32 + 16 + la] = acc1[v];
  }
  __syncthreads();

  const int kib = dh * KS_;
  for (int e = lane; e < 16 * 9 * KS_; e += 32) {
    const int dw = e % KS_;
    int r = e / KS_;
    const int si = r % 9;
    const int m  = r / 9;
    const int n  = iclip(w0 + m - R_, 0, W_ - KS_) + dw - c0;  // 0..21
    const float  dot = D[m * 32 + n];
    const size_t pix = (size_t)h * W_ + (w0 + m);
    const float  lp  = logp[((size_t)(b * 9 + si) * HWK + pix) * 49 + kib + dw];
    attn[((size_t)((b * G_ + g) * 9 + si) * HWK + pix) * 49 + kib + dw] =
        dot + lp;
  }
}

// ============================================================================
extern "C" void kernel_launch(void* const* d_in, const int* in_sizes, int n_in,
                              void* d_out, int out_size, void* d_ws,
                              size_t ws_size, hipStream_t stream) {
  const float* x     = (const float*)d_in[0];
  const float* sims  = (const float*)d_in[1];
  const int*   sinds = (const int*)d_in[2];
  const float* Wqk   = (const float*)d_in[3];
  const float* bqk   = (const float*)d_in[4];

  // workspace: qk (32768 x 1024 f32 = 128MB) then logp (B*9*H*W*49 f32 ~ 58MB)
  float* qk   = (float*)d_ws;
  float* logp = (float*)((char*)d_ws + (size_t)NPIX * QKN * sizeof(float));
  float* attn = (float*)d_out;

  // A: N-tiles fast in grid.x so same-M blocks are temporally adjacent (L2)
  qk_gemm_kernel<<<dim3(QKN / BN, NPIX / BM), 256, 0, stream>>>(x, Wqk, bqk, qk);

  // B: exactly B*H*W*49 = 1,605,632 threads = 6272 * 256
  logp_kernel<<<(B_ * HWK * 49) / 256, 256, 0, stream>>>(sims, sinds, logp);

  // C: B*G*H*(W/16)*7 = 57344 waves, 8 waves per block
  attn_kernel<<<(B_ * G_ * H_ * (W_ / 16) * 7) / 8, 256, 0, stream>>>(qk, logp,
                                                                      attn);
}